// SVGD_45208825757927
// MI455X (gfx1250) — compile-verified
//
#include <hip/hip_runtime.h>
#include <math.h>

typedef __attribute__((ext_vector_type(2))) float v2f;
typedef __attribute__((ext_vector_type(8))) float v8f;

#define N_PART 32
#define DIM 256
#define BATCH 8192
#define N_TRAIN 8192

// ---------------------------------------------------------------------------
// K1: z = W @ X^T (per batch column), sigmoid, dzdy, per-column stats.
// One wave32 owns one batch column; lane n = particle n. Stats reduced with
// __shfl_xor across the wave (exactly 32 particles == wave32).
// ---------------------------------------------------------------------------
__global__ __launch_bounds__(256) void k1_logits_stats(
    const float* __restrict__ X, const float* __restrict__ y,
    const float* __restrict__ W, float* __restrict__ dzdyT,
    float* __restrict__ sarr, float* __restrict__ llb, float* __restrict__ accb)
{
  __shared__ float Wl[N_PART * 257];   // pad 257: conflict-free column reads
  int t = threadIdx.x;
  for (int idx = t; idx < N_PART * DIM; idx += 256) {
    int n = idx >> 8, k = idx & 255;
    Wl[n * 257 + k] = W[idx];
  }
  __syncthreads();

  int n  = t & 31;          // particle = lane
  int wv = t >> 5;          // wave id within block (8 waves)
  int b  = blockIdx.x * 8 + wv;

  const float* xr = X + (size_t)b * DIM;
  const float* wr = Wl + n * 257;
  float acc = 0.0f;
  #pragma unroll 4
  for (int k = 0; k < DIM; k += 4) {
    float4 xv = *(const float4*)(xr + k);   // broadcast across the wave
    acc = fmaf(wr[k + 0], xv.x, acc);
    acc = fmaf(wr[k + 1], xv.y, acc);
    acc = fmaf(wr[k + 2], xv.z, acc);
    acc = fmaf(wr[k + 3], xv.w, acc);
  }
  float p  = 1.0f / (1.0f + expf(-acc));
  float yb = y[b];
  float denom = p - p * p + 1e-8f;
  float dz = (p * (1.0f - p)) * (yb - p) / denom;
  dzdyT[b * N_PART + n] = dz;             // [b][n] layout: coalesced store

  float sp = p, sd = dz, sd2 = dz * dz;
  for (int off = 16; off > 0; off >>= 1) {
    sp  += __shfl_xor(sp,  off);
    sd  += __shfl_xor(sd,  off);
    sd2 += __shfl_xor(sd2, off);
  }
  if (n == 0) {
    float yp = sp * (1.0f / 32.0f);
    float mm = sd * (1.0f / 32.0f);
    sarr[b] = sd2 - 32.0f * mm * mm;      // sum_n (dzdy - mean)^2
    llb[b]  = yb * logf(yp + 1e-3f) + (1.0f - yb) * logf(1.0f - yp + 1e-3f);
    accb[b] = ((yb > 0.5f) == (yp > 0.5f)) ? 1.0f : 0.0f;
  }
}

// ---------------------------------------------------------------------------
// K2: W_grads = (dzdy @ X)/BATCH * N_TRAIN - W  via V_WMMA_F32_16X16X4_F32.
// 32 tiles (2 x 16 of 16x16), K=8192 split across 8 waves, LDS reduction.
// ---------------------------------------------------------------------------
__global__ __launch_bounds__(256) void k2_wgrads_wmma(
    const float* __restrict__ dzdyT, const float* __restrict__ X,
    const float* __restrict__ W, float* __restrict__ Wg)
{
  int tile = blockIdx.x;                 // 0..31
  int n0 = (tile >> 4) << 4;
  int d0 = (tile & 15) << 4;
  int lane = threadIdx.x & 31;
  int wave = threadIdx.x >> 5;
  int m  = lane & 15;
  int hf = lane >> 4;                    // half-wave select (K=0/1 vs K=2/3)

  v8f acc = {};
  int bBeg = wave * (BATCH / 8);
  for (int b0 = bBeg; b0 < bBeg + BATCH / 8; b0 += 4) {
    int k0 = b0 + hf * 2;
    v2f a, bb;
    a[0]  = dzdyT[(size_t)(k0 + 0) * N_PART + n0 + m];   // A[m,k] = dzdy[n0+m, b]
    a[1]  = dzdyT[(size_t)(k0 + 1) * N_PART + n0 + m];
    bb[0] = X[(size_t)(k0 + 0) * DIM + d0 + m];          // B[k,n] = X[b, d0+n]
    bb[1] = X[(size_t)(k0 + 1) * DIM + d0 + m];
    acc = __builtin_amdgcn_wmma_f32_16x16x4_f32(
        false, a, false, bb, (short)0, acc, false, false);
  }

  __shared__ float red[8][256];
  #pragma unroll
  for (int r = 0; r < 8; ++r)
    red[wave][(r + hf * 8) * 16 + m] = acc[r];           // C/D layout -> (row,col)
  __syncthreads();

  int t = threadIdx.x;
  float v = 0.0f;
  #pragma unroll
  for (int w2 = 0; w2 < 8; ++w2) v += red[w2][t];
  int row = t >> 4, col = t & 15;
  int gn = n0 + row, gd = d0 + col;
  float dW = v / (float)BATCH;
  Wg[gn * DIM + gd] = dW * (float)N_TRAIN - W[gn * DIM + gd];
}

// ---------------------------------------------------------------------------
// K3: covReg = X^T diag(s) X / (N*B) + 0.01*I  via V_WMMA_F32_16X16X4_F32.
// 256 tiles of 16x16, K=8192 split across 8 waves, LDS reduction.
// (The einsum over the 268MB glW tensor collapses to this 1.07 GFLOP GEMM.)
// ---------------------------------------------------------------------------
__global__ __launch_bounds__(256) void k3_cov_wmma(
    const float* __restrict__ X, const float* __restrict__ sarr,
    float* __restrict__ covR)
{
  int tile = blockIdx.x;                 // 0..255
  int d0 = (tile >> 4) << 4;
  int e0 = (tile & 15) << 4;
  int lane = threadIdx.x & 31;
  int wave = threadIdx.x >> 5;
  int m  = lane & 15;
  int hf = lane >> 4;

  v8f acc = {};
  int bBeg = wave * (BATCH / 8);
  for (int b0 = bBeg; b0 < bBeg + BATCH / 8; b0 += 4) {
    int k0 = b0 + hf * 2;
    float s0 = sarr[k0 + 0];
    float s1 = sarr[k0 + 1];
    v2f a, bb;
    a[0]  = X[(size_t)(k0 + 0) * DIM + d0 + m];          // A[m,k] = X[b, d0+m]
    a[1]  = X[(size_t)(k0 + 1) * DIM + d0 + m];
    bb[0] = s0 * X[(size_t)(k0 + 0) * DIM + e0 + m];     // B[k,n] = s[b]*X[b, e0+n]
    bb[1] = s1 * X[(size_t)(k0 + 1) * DIM + e0 + m];
    acc = __builtin_amdgcn_wmma_f32_16x16x4_f32(
        false, a, false, bb, (short)0, acc, false, false);
  }

  __shared__ float red[8][256];
  #pragma unroll
  for (int r = 0; r < 8; ++r)
    red[wave][(r + hf * 8) * 16 + m] = acc[r];
  __syncthreads();

  int t = threadIdx.x;
  float v = 0.0f;
  #pragma unroll
  for (int w2 = 0; w2 < 8; ++w2) v += red[w2][t];
  int row = t >> 4, col = t & 15;
  int gd = d0 + row, ge = e0 + col;
  v *= 1.0f / ((float)N_PART * (float)BATCH);
  if (gd == ge) v += 0.01f;              // ridge folded into epilogue
  covR[gd * DIM + ge] = v;
}

// ---------------------------------------------------------------------------
// K4: H_inv = inv(covReg). SPD + 0.01 ridge -> Gauss-Jordan without pivoting.
// Whole 256x256 matrix lives in LDS (264KB of CDNA5's 320KB/WGP), 1024 thr.
// ---------------------------------------------------------------------------
#define GJS 257
__global__ __launch_bounds__(1024) void k4_invert(
    const float* __restrict__ A, float* __restrict__ Ainv)
{
  __shared__ float M[DIM * GJS];
  __shared__ float colk[DIM];
  int t = threadIdx.x;
  for (int idx = t; idx < DIM * DIM; idx += 1024) {
    int i = idx >> 8, j = idx & 255;
    M[i * GJS + j] = A[idx];
  }
  __syncthreads();
  for (int k = 0; k < DIM; ++k) {
    float pinv = 1.0f / M[k * GJS + k];
    __syncthreads();
    if (t < DIM) {
      M[k * GJS + t] = (t == k) ? pinv : M[k * GJS + t] * pinv;  // scale row k
      colk[t] = (t == k) ? 0.0f : M[t * GJS + k];                // snapshot col k
    }
    __syncthreads();
    for (int idx = t; idx < DIM * DIM; idx += 1024) {
      int i = idx >> 8, j = idx & 255;
      if (i == k) continue;
      float f = colk[i];
      float v;
      if (j == k) v = -f * pinv;
      else        v = M[i * GJS + j] - f * M[k * GJS + j];
      M[i * GJS + j] = v;
    }
    __syncthreads();
  }
  for (int idx = t; idx < DIM * DIM; idx += 1024) {
    int i = idx >> 8, j = idx & 255;
    Ainv[idx] = M[i * GJS + j];
  }
}

// ---------------------------------------------------------------------------
// K5: RBF kernel on W (median heuristic via exact in-LDS bitonic sort of the
// 1024 pairwise sq-distances) + svgd_grad = (kxy@W_grads + dxkxy)/N_PART.
// ---------------------------------------------------------------------------
__global__ __launch_bounds__(1024) void k5_svgd(
    const float* __restrict__ W, const float* __restrict__ Wg,
    float* __restrict__ outSvgd)
{
  __shared__ float Wl[N_PART * 257];
  __shared__ float pdA[1024];
  __shared__ float srt[1024];
  __shared__ float kxyA[1024];
  __shared__ float sq[N_PART];
  __shared__ float sumk[N_PART];
  int t = threadIdx.x;

  for (int idx = t; idx < N_PART * DIM; idx += 1024) {
    int n = idx >> 8, k = idx & 255;
    Wl[n * 257 + k] = W[idx];
  }
  __syncthreads();
  if (t < N_PART) {
    float a = 0.0f;
    const float* wr = Wl + t * 257;
    for (int k = 0; k < DIM; ++k) a = fmaf(wr[k], wr[k], a);
    sq[t] = a;
  }
  __syncthreads();
  {
    int i = t >> 5, j = t & 31;                  // each thread one (i,j) pair
    const float* wi = Wl + i * 257;
    const float* wj = Wl + j * 257;
    float d = 0.0f;
    for (int k = 0; k < DIM; ++k) d = fmaf(wi[k], wj[k], d);
    float pv = sq[i] - 2.0f * d + sq[j];
    pdA[t] = pv;
    srt[t] = pv;
  }
  __syncthreads();
  // bitonic sort (1024 elements, 1024 threads) for the exact median
  for (int kk = 2; kk <= 1024; kk <<= 1) {
    for (int j = kk >> 1; j > 0; j >>= 1) {
      int ixj = t ^ j;
      if (ixj > t) {
        float a = srt[t], b = srt[ixj];
        bool up = ((t & kk) == 0);
        if (up ? (a > b) : (a < b)) { srt[t] = b; srt[ixj] = a; }
      }
      __syncthreads();
    }
  }
  float h = (0.5f * (srt[511] + srt[512])) / logf(33.0f);  // median/log(n+1)
  kxyA[t] = expf(-pdA[t] / h);
  __syncthreads();
  if (t < N_PART) {
    float a = 0.0f;
    for (int j = 0; j < N_PART; ++j) a += kxyA[t * 32 + j];
    sumk[t] = a;
  }
  __syncthreads();
  float twoh = 2.0f / h;
  for (int idx = t; idx < N_PART * DIM; idx += 1024) {
    int i = idx >> 8, d = idx & 255;
    float a1 = 0.0f, a2 = 0.0f;
    #pragma unroll 8
    for (int j = 0; j < N_PART; ++j) {
      float kv = kxyA[i * 32 + j];
      a1 = fmaf(kv, Wg[j * DIM + d], a1);      // kxy @ W_grads
      a2 = fmaf(kv, Wl[j * 257 + d], a2);      // kxy @ W
    }
    float dx = (Wl[i * 257 + d] * sumk[i] - a2) * twoh;
    outSvgd[idx] = (a1 + dx) * (1.0f / (float)N_PART);
  }
}

// ---------------------------------------------------------------------------
// K6: kfac_grad = svgd_grad @ H_inv   (32x256 @ 256x256, coalesced rank-1s)
// ---------------------------------------------------------------------------
__global__ __launch_bounds__(256) void k6_kfac(
    const float* __restrict__ svgd, const float* __restrict__ Hinv,
    float* __restrict__ outK)
{
  __shared__ float row[DIM];
  int i = blockIdx.x, t = threadIdx.x;
  row[t] = svgd[i * DIM + t];
  __syncthreads();
  float acc = 0.0f;
  #pragma unroll 4
  for (int d = 0; d < DIM; ++d)
    acc = fmaf(row[d], Hinv[d * DIM + t], acc);
  outK[i * DIM + t] = acc;
}

// ---------------------------------------------------------------------------
// K7: deterministic tree reduction of ll/acc over the batch.
// ---------------------------------------------------------------------------
__global__ __launch_bounds__(256) void k7_stats(
    const float* __restrict__ llb, const float* __restrict__ accb,
    float* __restrict__ out)
{
  __shared__ float sl[256], sa[256];
  int t = threadIdx.x;
  float a = 0.0f, b = 0.0f;
  for (int i = t; i < BATCH; i += 256) { a += llb[i]; b += accb[i]; }
  sl[t] = a; sa[t] = b;
  __syncthreads();
  for (int s = 128; s > 0; s >>= 1) {
    if (t < s) { sl[t] += sl[t + s]; sa[t] += sa[t + s]; }
    __syncthreads();
  }
  if (t == 0) {
    out[0] = sl[0] / (float)BATCH;
    out[1] = sa[0] / (float)BATCH;
  }
}

// ---------------------------------------------------------------------------
extern "C" void kernel_launch(void* const* d_in, const int* in_sizes, int n_in,
                              void* d_out, int out_size, void* d_ws, size_t ws_size,
                              hipStream_t stream)
{
  (void)in_sizes; (void)n_in; (void)out_size; (void)ws_size;
  const float* X = (const float*)d_in[0];   // [8192, 256]
  const float* y = (const float*)d_in[1];   // [8192]
  const float* W = (const float*)d_in[2];   // [32, 256]
  float* out = (float*)d_out;               // ll, acc, svgd[32*256], kfac[32*256]
  float* ws  = (float*)d_ws;

  float* dzdyT = ws;                        // [8192][32]
  float* sarr  = dzdyT + (size_t)BATCH * N_PART;  // [8192]
  float* llb   = sarr + BATCH;                    // [8192]
  float* accb  = llb + BATCH;                     // [8192]
  float* Wg    = accb + BATCH;                    // [32][256]
  float* covR  = Wg + N_PART * DIM;               // [256][256]
  float* Hinv  = covR + DIM * DIM;                // [256][256]

  float* outLL   = out;
  float* outSvgd = out + 2;
  float* outKfac = out + 2 + N_PART * DIM;

  k1_logits_stats<<<BATCH / 8, 256, 0, stream>>>(X, y, W, dzdyT, sarr, llb, accb);
  k2_wgrads_wmma<<<32, 256, 0, stream>>>(dzdyT, X, W, Wg);
  k3_cov_wmma<<<256, 256, 0, stream>>>(X, sarr, covR);
  k4_invert<<<1, 1024, 0, stream>>>(covR, Hinv);
  k5_svgd<<<1, 1024, 0, stream>>>(W, Wg, outSvgd);
  k6_kfac<<<32, 256, 0, stream>>>(outSvgd, Hinv, outKfac);
  k7_stats<<<1, 256, 0, stream>>>(llb, accb, outLL);
}